// GNN_90134183674353
// MI455X (gfx1250) — compile-verified
//
#include <hip/hip_runtime.h>
#include <math.h>

typedef float v2f __attribute__((ext_vector_type(2)));
typedef float v8f __attribute__((ext_vector_type(8)));

#define DIM 64

// ---------------------------------------------------------------- fill -inf
__global__ __launch_bounds__(256) void sage_fill_neginf(float4* __restrict__ p, int n4) {
    int i = blockIdx.x * 256 + threadIdx.x;
    if (i < n4) {
        float ni = -__builtin_inff();
        p[i] = make_float4(ni, ni, ni, ni);
    }
}

// ------------------------------------------------- scatter-max over edges
// one thread per (edge, group-of-4 features): float4 gather + 4 atomic fmax
__global__ __launch_bounds__(256) void sage_scatter_max(
    const float* __restrict__ h,
    const int* __restrict__ src,
    const int* __restrict__ dst,
    float* __restrict__ agg,
    int n_edges)
{
    int idx = blockIdx.x * 256 + threadIdx.x;
    if (idx >= n_edges * 16) return;
    int e = idx >> 4;
    int q = (idx & 15) << 2;               // feature offset 0..60 step 4
    int s = src[e];
    int d = dst[e];
    const float4 v = *(const float4*)(h + (size_t)s * DIM + q);
    float* ap = agg + (size_t)d * DIM + q;
    atomicMax(ap + 0, v.x);                 // global_atomic_max_num_f32
    atomicMax(ap + 1, v.y);
    atomicMax(ap + 2, v.z);
    atomicMax(ap + 3, v.w);
}

// --------------------------------------------------- fused dual-GEMM + bias
// out = relu?( clean(agg) @ Wl^T + bl + h @ Wr^T )
// block = 256 threads = 8 waves; each wave owns 16 rows x 64 cols.
// Weights staged in LDS with K-pair interleaving so each WMMA B fragment is
// one aligned ds_load_b64: sW2[(k/2)*128 + j*2 + (k&1)] = W[j*64 + k].
__global__ __launch_bounds__(256) void sage_gemm_wmma(
    const float* __restrict__ agg,   // [N,64], -inf marks "no in-edges"
    const float* __restrict__ h,     // [N,64]
    const float* __restrict__ Wl,    // [64,64] row j = output dim, col k = input dim
    const float* __restrict__ bl,    // [64]
    const float* __restrict__ Wr,    // [64,64]
    float* __restrict__ out,         // [N,64]
    int n_nodes, int do_relu)
{
    __shared__ __align__(16) float sWl[DIM * DIM];
    __shared__ __align__(16) float sWr[DIM * DIM];
    __shared__ float sb[DIM];

    for (int i = threadIdx.x; i < DIM * DIM; i += 256) {
        int j = i >> 6, k = i & 63;
        int d = (k >> 1) * 128 + j * 2 + (k & 1);   // K-pair interleaved transpose
        sWl[d] = Wl[i];
        sWr[d] = Wr[i];
    }
    if (threadIdx.x < DIM) sb[threadIdx.x] = bl[threadIdx.x];
    __syncthreads();

    const int wave = threadIdx.x >> 5;       // 0..7
    const int lane = threadIdx.x & 31;
    const int half = lane >> 4;              // 0 | 1
    const int l16  = lane & 15;

    const int m0 = (blockIdx.x * 8 + wave) * 16;   // 16-row tile base
    if (m0 >= n_nodes) return;                     // uniform per-wave exit

    // clamp per-lane source row (stores are guarded; WMMA keeps EXEC all-1s)
    int arow = m0 + l16;
    if (arow >= n_nodes) arow = n_nodes - 1;
    const float2* aggRow = (const float2*)(agg + (size_t)arow * DIM);
    const float2* hRow   = (const float2*)(h   + (size_t)arow * DIM);

    const v2f* pWl = (const v2f*)sWl;   // [32 pair-rows][64 cols]
    const v2f* pWr = (const v2f*)sWr;

    // accumulators seeded with bias (C-matrix col = j0 + l16)
    v8f acc[4];
    for (int t = 0; t < 4; ++t) {
        float b = sb[t * 16 + l16];
        acc[t] = (v8f){b, b, b, b, b, b, b, b};
    }

    const float ninf = -__builtin_inff();
    for (int k0 = 0; k0 < DIM; k0 += 4) {
        const int kb = k0 + half * 2;        // this lane's K pair base (even)
        const int p  = kb >> 1;              // pair-row index
        float2 av = aggRow[p];
        float2 hv = hRow[p];
        if (av.x == ninf) av.x = 0.0f;       // PyG: no in-edges -> 0
        if (av.y == ninf) av.y = 0.0f;
        v2f aA; aA.x = av.x; aA.y = av.y;
        v2f aH; aH.x = hv.x; aH.y = hv.y;
        #pragma unroll
        for (int t = 0; t < 4; ++t) {
            const int j0 = t * 16;
            v2f bL = pWl[p * 64 + j0 + l16]; // single aligned ds_load_b64
            v2f bR = pWr[p * 64 + j0 + l16];
            acc[t] = __builtin_amdgcn_wmma_f32_16x16x4_f32(
                false, aA, false, bL, (short)0, acc[t], false, false);
            acc[t] = __builtin_amdgcn_wmma_f32_16x16x4_f32(
                false, aH, false, bR, (short)0, acc[t], false, false);
        }
    }

    // D layout: VGPR v -> row m0 + 8*half + v, col j0 + l16
    #pragma unroll
    for (int t = 0; t < 4; ++t) {
        const int j0 = t * 16;
        #pragma unroll
        for (int v = 0; v < 8; ++v) {
            int row = m0 + half * 8 + v;
            if (row < n_nodes) {
                float val = acc[t][v];
                if (do_relu) val = fmaxf(val, 0.0f);
                out[(size_t)row * DIM + j0 + l16] = val;
            }
        }
    }
}

// ---------------------------------------------------------------- driver
extern "C" void kernel_launch(void* const* d_in, const int* in_sizes, int n_in,
                              void* d_out, int out_size, void* d_ws, size_t ws_size,
                              hipStream_t stream) {
    const float* x    = (const float*)d_in[0];   // [N,64]
    const int*   edge = (const int*)d_in[1];     // [2,E]
    const float* Wl   = (const float*)d_in[2];   // [6,64,64]
    const float* bl   = (const float*)d_in[3];   // [6,64]
    const float* Wr   = (const float*)d_in[4];   // [6,64,64]
    float* out = (float*)d_out;

    const int N = in_sizes[0] / DIM;
    const int E = in_sizes[1] / 2;
    const int n_layers = in_sizes[3] / DIM;

    const int* src = edge;
    const int* dst = edge + E;

    float* ws_h = (float*)d_ws;                    // [N,64] ping buffer
    float* agg  = ws_h + (size_t)N * DIM;          // [N,64] aggregation buffer

    const int fill_n4   = N * (DIM / 4);
    const int fill_blk  = (fill_n4 + 255) / 256;
    const int scat_tot  = E * 16;
    const int scat_blk  = (scat_tot + 255) / 256;
    const int gemm_blk  = (N + 127) / 128;         // 8 waves x 16 rows per block

    const float* hin = x;
    for (int layer = 0; layer < n_layers; ++layer) {
        // even layers -> ws buffer, odd layers -> d_out; last layer (5, odd) ends in d_out
        float* hout = (layer & 1) ? out : ws_h;

        sage_fill_neginf<<<fill_blk, 256, 0, stream>>>((float4*)agg, fill_n4);
        sage_scatter_max<<<scat_blk, 256, 0, stream>>>(hin, src, dst, agg, E);
        sage_gemm_wmma<<<gemm_blk, 256, 0, stream>>>(
            agg, hin,
            Wl + (size_t)layer * DIM * DIM,
            bl + (size_t)layer * DIM,
            Wr + (size_t)layer * DIM * DIM,
            hout, N, (layer < n_layers - 1) ? 1 : 0);

        hin = hout;
    }
}